// AvgNet_37795712205092
// MI455X (gfx1250) — compile-verified
//
#include <hip/hip_runtime.h>
#include <hip/hip_bf16.h>
#include <math.h>

#define D_MODEL 1024
#define SL      2048
#define BS      8

typedef float v2f __attribute__((ext_vector_type(2)));
typedef float v8f __attribute__((ext_vector_type(8)));

// ---------------------------------------------------------------------------
// K1: ksumraw[b,d] = sum_j k[b,j,d]       grid(4,8) x 256
// ---------------------------------------------------------------------------
__global__ __launch_bounds__(256)
void colsum_kernel(const float* __restrict__ k, float* __restrict__ ksumraw) {
    const int b = blockIdx.y;
    const int d = blockIdx.x * 256 + threadIdx.x;
    const float* kb = k + (size_t)b * SL * D_MODEL + d;
    float acc = 0.f;
    for (int j = 0; j < SL; ++j) acc += kb[(size_t)j * D_MODEL];
    ksumraw[b * D_MODEL + d] = acc;
}

// ---------------------------------------------------------------------------
// K2/K6: y[b,n] = x[b,:] . W[n,:] + bias[n]*bias_scale   (one wave32 / output)
// grid 1024 x 256 (8 waves/block, 8192 outputs)
// ---------------------------------------------------------------------------
__global__ __launch_bounds__(256)
void matvec_kernel(const float* __restrict__ x, const float* __restrict__ W,
                   const float* __restrict__ bias, float bias_scale,
                   float* __restrict__ y) {
    const int lane = threadIdx.x & 31;
    const int wave = threadIdx.x >> 5;
    const int g = blockIdx.x * 8 + wave;     // 0..8191
    const int b = g >> 10;
    const int n = g & (D_MODEL - 1);
    const float* xb = x + b * D_MODEL;
    const float* wr = W + (size_t)n * D_MODEL;
    float acc = 0.f;
    for (int d = lane; d < D_MODEL; d += 32) acc += xb[d] * wr[d];
    #pragma unroll
    for (int m = 1; m < 32; m <<= 1) acc += __shfl_xor(acc, m, 32);
    if (lane == 0) y[g] = acc + bias[n] * bias_scale;
}

// ---------------------------------------------------------------------------
// K3 (main WMMA kernel)
// Each wave owns a 16-row tile and walks N in PAIRS of 16-col tiles (sharing
// the A-operand loads), with the K loop software-pipelined in 16-wide chunks
// held in registers (double buffered):  24 b64 loads cover 16 WMMAs/chunk.
// ---------------------------------------------------------------------------
struct KChunk {
    v2f aq[4], ak[4];                 // A rows (shared by both N-tiles)
    v2f bq0[4], bk0[4], bq1[4], bk1[4];
};

__device__ __forceinline__ void load_chunk(KChunk& c,
        const float* __restrict__ qrow, const float* __restrict__ krow,
        const float* __restrict__ wq0, const float* __restrict__ wk0,
        const float* __restrict__ wq1, const float* __restrict__ wk1,
        int kbase, int half) {
    #pragma unroll
    for (int s = 0; s < 4; ++s) {
        const int kk = kbase + 4 * s + 2 * half;
        c.aq[s]  = *(const v2f*)(qrow + kk);
        c.ak[s]  = *(const v2f*)(krow + kk);
        c.bq0[s] = *(const v2f*)(wq0 + kk);
        c.bk0[s] = *(const v2f*)(wk0 + kk);
        c.bq1[s] = *(const v2f*)(wq1 + kk);
        c.bk1[s] = *(const v2f*)(wk1 + kk);
    }
}

__device__ __forceinline__ void mma_chunk(const KChunk& c,
        v8f& qp0, v8f& kp0, v8f& qp1, v8f& kp1) {
    #pragma unroll
    for (int s = 0; s < 4; ++s) {
        qp0 = __builtin_amdgcn_wmma_f32_16x16x4_f32(false, c.aq[s], false, c.bq0[s], (short)0, qp0, false, false);
        kp0 = __builtin_amdgcn_wmma_f32_16x16x4_f32(false, c.ak[s], false, c.bk0[s], (short)0, kp0, false, false);
        qp1 = __builtin_amdgcn_wmma_f32_16x16x4_f32(false, c.aq[s], false, c.bq1[s], (short)0, qp1, false, false);
        kp1 = __builtin_amdgcn_wmma_f32_16x16x4_f32(false, c.ak[s], false, c.bk1[s], (short)0, kp1, false, false);
    }
}

__global__ __launch_bounds__(128)
void proj_score_kernel(const float* __restrict__ q, const float* __restrict__ k,
                       const float* __restrict__ Wq, const float* __restrict__ bq,
                       const float* __restrict__ Wk, const float* __restrict__ bk,
                       const float* __restrict__ ksum_p, float* __restrict__ s_out) {
    const int lane = threadIdx.x & 31;
    const int wave = threadIdx.x >> 5;
    const int tile = blockIdx.x * 4 + wave;   // 0..1023
    const int grow0 = tile * 16;              // first global row of this tile
    const int b = grow0 / SL;                 // tiles never cross a batch
    const int lo = lane & 15;
    const int half = lane >> 4;

    const float* qrow = q + (size_t)(grow0 + lo) * D_MODEL;   // A operand rows
    const float* krow = k + (size_t)(grow0 + lo) * D_MODEL;
    const float* ksb  = ksum_p + b * D_MODEL;

    v8f dotAcc = {};   // partials of qp_i . kp_i    (summed over ALL columns)
    v8f qsAcc  = {};   // partials of qp_i . ksum

    for (int n0 = 0; n0 < D_MODEL; n0 += 32) {
        const float* wq0 = Wq + (size_t)(n0 + lo) * D_MODEL;
        const float* wk0 = Wk + (size_t)(n0 + lo) * D_MODEL;
        const float* wq1 = wq0 + 16 * D_MODEL;
        const float* wk1 = wk0 + 16 * D_MODEL;

        // C/D layout: VGPR r, lane l -> (M = r + 8*(l>>4), N = l&15); bias per-N
        const float bq0v = bq[n0 + lo],      bk0v = bk[n0 + lo];
        const float bq1v = bq[n0 + 16 + lo], bk1v = bk[n0 + 16 + lo];
        v8f qp0, kp0, qp1, kp1;
        #pragma unroll
        for (int r = 0; r < 8; ++r) {
            qp0[r] = bq0v; kp0[r] = bk0v;
            qp1[r] = bq1v; kp1[r] = bk1v;
        }

        KChunk ca, cb;
        load_chunk(ca, qrow, krow, wq0, wk0, wq1, wk1, 0, half);
        for (int kc = 0; kc < D_MODEL; kc += 32) {
            // stream-prefetch A rows ahead of the pipeline (global_prefetch_b8)
            __builtin_prefetch(qrow + kc + 128, 0, 1);
            __builtin_prefetch(krow + kc + 128, 0, 1);
            load_chunk(cb, qrow, krow, wq0, wk0, wq1, wk1, kc + 16, half);
            mma_chunk(ca, qp0, kp0, qp1, kp1);          // overlaps cb's loads
            if (kc + 32 < D_MODEL)
                load_chunk(ca, qrow, krow, wq0, wk0, wq1, wk1, kc + 32, half);
            mma_chunk(cb, qp0, kp0, qp1, kp1);          // overlaps ca's loads
        }

        const float ks0 = ksb[n0 + lo];
        const float ks1 = ksb[n0 + 16 + lo];
        #pragma unroll
        for (int r = 0; r < 8; ++r) {
            dotAcc[r] += qp0[r] * kp0[r] + qp1[r] * kp1[r];
            qsAcc[r]  += qp0[r] * ks0   + qp1[r] * ks1;
        }
    }

    // Reduce over N: sum across the 16 lanes of each half.
    const float inv_sqrt_d = 0.03125f;        // 1/sqrt(1024)
    #pragma unroll
    for (int r = 0; r < 8; ++r) {
        float dv = dotAcc[r];
        float qv = qsAcc[r];
        #pragma unroll
        for (int m = 1; m <= 8; m <<= 1) {
            dv += __shfl_xor(dv, m, 32);
            qv += __shfl_xor(qv, m, 32);
        }
        if (lo == 0) {                        // lanes 0 and 16 -> rows r and r+8
            const int row = grow0 + r + 8 * half;
            s_out[row] = (qv - dv) * inv_sqrt_d;
        }
    }
}

// ---------------------------------------------------------------------------
// K4: softmax over s[b,:] -> w[b,:]; also zero-init pooled.  grid 8 x 256
// ---------------------------------------------------------------------------
__global__ __launch_bounds__(256)
void softmax_kernel(const float* __restrict__ s, float* __restrict__ w,
                    float* __restrict__ pooled) {
    const int b = blockIdx.x;
    const int t = threadIdx.x;
    __shared__ float red[256];

    for (int i = t; i < D_MODEL; i += 256) pooled[b * D_MODEL + i] = 0.f;

    const float* sb = s + b * SL;
    float m = -INFINITY;
    for (int i = t; i < SL; i += 256) m = fmaxf(m, sb[i]);
    red[t] = m; __syncthreads();
    for (int o = 128; o > 0; o >>= 1) {
        if (t < o) red[t] = fmaxf(red[t], red[t + o]);
        __syncthreads();
    }
    m = red[0]; __syncthreads();

    float sum = 0.f;
    for (int i = t; i < SL; i += 256) sum += __expf(sb[i] - m);
    red[t] = sum; __syncthreads();
    for (int o = 128; o > 0; o >>= 1) {
        if (t < o) red[t] += red[t + o];
        __syncthreads();
    }
    const float inv = 1.f / red[0];

    for (int i = t; i < SL; i += 256) w[b * SL + i] = __expf(sb[i] - m) * inv;
}

// ---------------------------------------------------------------------------
// K5: pooled[b,d] += sum_j w[b,j]*v[b,j,d]    grid(16,4,8) x 256, f32 atomics
// ---------------------------------------------------------------------------
__global__ __launch_bounds__(256)
void pool_kernel(const float* __restrict__ v, const float* __restrict__ w,
                 float* __restrict__ pooled) {
    const int b  = blockIdx.z;
    const int d  = blockIdx.y * 256 + threadIdx.x;
    const int j0 = blockIdx.x * (SL / 16);    // 128-row slabs
    const float* vb = v + ((size_t)b * SL + j0) * D_MODEL + d;
    const float* wb = w + b * SL + j0;
    float acc = 0.f;
    for (int j = 0; j < SL / 16; ++j) acc += wb[j] * vb[(size_t)j * D_MODEL];
    atomicAdd(&pooled[b * D_MODEL + d], acc);
}

// ---------------------------------------------------------------------------
extern "C" void kernel_launch(void* const* d_in, const int* in_sizes, int n_in,
                              void* d_out, int out_size, void* d_ws, size_t ws_size,
                              hipStream_t stream) {
    const float* q  = (const float*)d_in[0];
    const float* k  = (const float*)d_in[1];
    const float* v  = (const float*)d_in[2];
    const float* Wq = (const float*)d_in[3];
    const float* bq = (const float*)d_in[4];
    const float* Wk = (const float*)d_in[5];
    const float* bk = (const float*)d_in[6];
    const float* Wo = (const float*)d_in[7];
    const float* bo = (const float*)d_in[8];
    float* out = (float*)d_out;

    float* ws       = (float*)d_ws;
    float* ksumraw  = ws;                     // 8*1024
    float* ksum_p   = ws + 8192;              // 8*1024
    float* s        = ws + 16384;             // 8*2048
    float* w        = ws + 32768;             // 8*2048
    float* pooled   = ws + 49152;             // 8*1024

    colsum_kernel<<<dim3(4, 8), 256, 0, stream>>>(k, ksumraw);
    matvec_kernel<<<1024, 256, 0, stream>>>(ksumraw, Wk, bk, (float)SL, ksum_p);
    proj_score_kernel<<<256, 128, 0, stream>>>(q, k, Wq, bq, Wk, bk, ksum_p, s);
    softmax_kernel<<<8, 256, 0, stream>>>(s, w, pooled);
    pool_kernel<<<dim3(16, 4, 8), 256, 0, stream>>>(v, w, pooled);
    matvec_kernel<<<1024, 256, 0, stream>>>(pooled, Wo, bo, 1.0f, out);
}